// DistanceContrastiveLoss_69320772157802
// MI455X (gfx1250) — compile-verified
//
#include <hip/hip_runtime.h>
#include <math.h>

typedef __attribute__((ext_vector_type(2))) float v2f;
typedef __attribute__((ext_vector_type(8))) float v8f;

#define N_PTS 8192
#define DIMS  256

// ---------------------------------------------------------------------------
// Kernel 0: zero the row/col sum accumulators (harness poisons ws with 0xAA)
// ---------------------------------------------------------------------------
__global__ void dcl_zero_kernel(float* __restrict__ p, int n) {
    int i = blockIdx.x * blockDim.x + threadIdx.x;
    if (i < n) p[i] = 0.0f;
}

// ---------------------------------------------------------------------------
// Kernel 1: squared L2 norms of each row of cond (c2) and sol (s2).
// One wave32 per row; lane sums 8 strided elements, shfl_xor tree reduce.
// ---------------------------------------------------------------------------
__global__ void dcl_norms_kernel(const float* __restrict__ c,
                                 const float* __restrict__ s,
                                 float* __restrict__ c2,
                                 float* __restrict__ s2) {
    int gwave = (blockIdx.x * blockDim.x + threadIdx.x) >> 5;
    int lane  = threadIdx.x & 31;
    const float* src;
    float* dst;
    int row;
    if (gwave < N_PTS) { src = c; dst = c2; row = gwave; }
    else               { src = s; dst = s2; row = gwave - N_PTS; }
    const float* p = src + (size_t)row * DIMS;
    float acc = 0.0f;
#pragma unroll
    for (int k = lane; k < DIMS; k += 32) { float v = p[k]; acc += v * v; }
#pragma unroll
    for (int m = 16; m; m >>= 1) acc += __shfl_xor(acc, m, 32);
    if (lane == 0) dst[row] = acc;
}

// ---------------------------------------------------------------------------
// Kernel 2: tiled fp32 WMMA GEMM (c @ s^T) fused with sim/exp epilogue and
// row/col sum-of-exp accumulation (shift-0 logsumexp; sim <= 0 so safe).
// Block = 8 waves, tile 128 rows x 64 cols; wave tile 16x64 (4 accumulators).
// ---------------------------------------------------------------------------
__global__ __launch_bounds__(256)
void dcl_simloss_kernel(const float* __restrict__ C,
                        const float* __restrict__ S,
                        const float* __restrict__ c2,
                        const float* __restrict__ s2,
                        const float* __restrict__ temp,
                        float* __restrict__ rowsum,
                        float* __restrict__ colsum,
                        float* __restrict__ diag) {
    __shared__ float colAcc[64];

    const int tid  = threadIdx.x;
    const int wave = tid >> 5;
    const int lane = tid & 31;
    const int half = lane >> 4;   // 0: lanes 0-15, 1: lanes 16-31
    const int l16  = lane & 15;

    if (tid < 64) colAcc[tid] = 0.0f;
    __syncthreads();

    const int blockRow = blockIdx.y * 128;
    const int blockCol = blockIdx.x * 64;

    // A fragment: lane holds row M=l16, K pair selected by half (ISA 16x4 layout)
    const float* aPtr = C + (size_t)(blockRow + wave * 16 + l16) * DIMS + 2 * half;
    // B fragments (4 column tiles of 16): lane holds col N=l16, K pair by half
    const float* b0 = S + (size_t)(blockCol +  0 + l16) * DIMS + 2 * half;
    const float* b1 = S + (size_t)(blockCol + 16 + l16) * DIMS + 2 * half;
    const float* b2 = S + (size_t)(blockCol + 32 + l16) * DIMS + 2 * half;
    const float* b3 = S + (size_t)(blockCol + 48 + l16) * DIMS + 2 * half;

    v8f acc0 = {}, acc1 = {}, acc2 = {}, acc3 = {};

    for (int kc = 0; kc < DIMS; kc += 64) {
        if (kc + 64 < DIMS) {   // uniform branch: EXEC unchanged
            __builtin_prefetch(aPtr + kc + 64, 0, 0);
            __builtin_prefetch(b0   + kc + 64, 0, 0);
            __builtin_prefetch(b1   + kc + 64, 0, 0);
            __builtin_prefetch(b2   + kc + 64, 0, 0);
            __builtin_prefetch(b3   + kc + 64, 0, 0);
        }
#pragma unroll
        for (int kk = 0; kk < 64; kk += 4) {
            const int k0 = kc + kk;
            v2f a  = *(const v2f*)(aPtr + k0);
            v2f v0 = *(const v2f*)(b0 + k0);
            v2f v1 = *(const v2f*)(b1 + k0);
            v2f v2 = *(const v2f*)(b2 + k0);
            v2f v3 = *(const v2f*)(b3 + k0);
            acc0 = __builtin_amdgcn_wmma_f32_16x16x4_f32(false, a, false, v0,
                                                         (short)0, acc0, false, false);
            acc1 = __builtin_amdgcn_wmma_f32_16x16x4_f32(false, a, false, v1,
                                                         (short)0, acc1, false, false);
            acc2 = __builtin_amdgcn_wmma_f32_16x16x4_f32(false, a, false, v2,
                                                         (short)0, acc2, false, false);
            acc3 = __builtin_amdgcn_wmma_f32_16x16x4_f32(false, a, false, v3,
                                                         (short)0, acc3, false, false);
        }
    }

    // ---- epilogue: sim = -sqrt(max(c2+s2-2*dot,0)) * e^t ; accumulate exp ----
    const float eT = __expf(temp[0]);

    float c2r[8];
#pragma unroll
    for (int r = 0; r < 8; ++r)
        c2r[r] = c2[blockRow + wave * 16 + 8 * half + r];
    float s2c[4];
#pragma unroll
    for (int t = 0; t < 4; ++t)
        s2c[t] = s2[blockCol + t * 16 + l16];

    float rs[8] = {0.f, 0.f, 0.f, 0.f, 0.f, 0.f, 0.f, 0.f};
    float cs[4] = {0.f, 0.f, 0.f, 0.f};
    v8f accs[4] = {acc0, acc1, acc2, acc3};

#pragma unroll
    for (int t = 0; t < 4; ++t) {
        const int gc = blockCol + t * 16 + l16;
#pragma unroll
        for (int r = 0; r < 8; ++r) {
            // C/D layout: element (vgpr r, lane) -> row M = r + 8*half, col N = l16
            const int gr = blockRow + wave * 16 + 8 * half + r;
            float d2  = c2r[r] + s2c[t] - 2.0f * accs[t][r];
            float sim = -sqrtf(fmaxf(d2, 0.0f)) * eT;
            if (gr == gc) diag[gr] = sim;
            float e = __expf(sim);
            rs[r] += e;
            cs[t] += e;
        }
    }

    // row sums: reduce across the 16 lanes of each half (masks 1,2,4,8)
#pragma unroll
    for (int r = 0; r < 8; ++r) {
        float v = rs[r];
        v += __shfl_xor(v, 1, 32);
        v += __shfl_xor(v, 2, 32);
        v += __shfl_xor(v, 4, 32);
        v += __shfl_xor(v, 8, 32);
        if (l16 == 0)
            atomicAdd(&rowsum[blockRow + wave * 16 + 8 * half + r], v);
    }

    // col sums: halves hold the same column -> xor 16, then LDS aggregate across waves
#pragma unroll
    for (int t = 0; t < 4; ++t) {
        float v = cs[t] + __shfl_xor(cs[t], 16, 32);
        if (half == 0)
            atomicAdd(&colAcc[t * 16 + l16], v);   // ds_add_f32
    }
    __syncthreads();
    if (tid < 64)
        atomicAdd(&colsum[blockCol + tid], colAcc[tid]);
}

// ---------------------------------------------------------------------------
// Kernel 3: loss = 0.5 * mean( (-diag + log(rowsum)) + (-diag + log(colsum)) )
// ---------------------------------------------------------------------------
__global__ void dcl_finalize_kernel(const float* __restrict__ rowsum,
                                    const float* __restrict__ colsum,
                                    const float* __restrict__ diag,
                                    float* __restrict__ out) {
    __shared__ float red[8];
    float acc = 0.0f;
    for (int i = threadIdx.x; i < N_PTS; i += 256) {
        float d = diag[i];
        acc += (-d + logf(rowsum[i]));
        acc += (-d + logf(colsum[i]));
    }
#pragma unroll
    for (int m = 16; m; m >>= 1) acc += __shfl_xor(acc, m, 32);
    int lane = threadIdx.x & 31, wave = threadIdx.x >> 5;
    if (lane == 0) red[wave] = acc;
    __syncthreads();
    if (threadIdx.x == 0) {
        float t = 0.0f;
#pragma unroll
        for (int w = 0; w < 8; ++w) t += red[w];
        out[0] = 0.5f * t / (float)N_PTS;
    }
}

// ---------------------------------------------------------------------------
extern "C" void kernel_launch(void* const* d_in, const int* in_sizes, int n_in,
                              void* d_out, int out_size, void* d_ws, size_t ws_size,
                              hipStream_t stream) {
    const float* cond = (const float*)d_in[0];
    const float* sol  = (const float*)d_in[1];
    const float* temp = (const float*)d_in[2];

    float* ws     = (float*)d_ws;
    float* c2     = ws;                 // [8192]
    float* s2     = ws + 1 * N_PTS;     // [8192]
    float* rowsum = ws + 2 * N_PTS;     // [8192]
    float* colsum = ws + 3 * N_PTS;     // [8192]  (contiguous with rowsum)
    float* diag   = ws + 4 * N_PTS;     // [8192]

    dcl_zero_kernel<<<(2 * N_PTS + 255) / 256, 256, 0, stream>>>(rowsum, 2 * N_PTS);
    dcl_norms_kernel<<<(2 * N_PTS) / 8, 256, 0, stream>>>(cond, sol, c2, s2);

    dim3 grid(N_PTS / 64, N_PTS / 128);   // (128, 64) blocks of 256 threads
    dcl_simloss_kernel<<<grid, 256, 0, stream>>>(cond, sol, c2, s2, temp,
                                                 rowsum, colsum, diag);

    dcl_finalize_kernel<<<1, 256, 0, stream>>>(rowsum, colsum, diag, (float*)d_out);
}